// Block_57552561766797
// MI455X (gfx1250) — compile-verified
//
#include <hip/hip_runtime.h>

// ---------------------------------------------------------------------------
// GNN block for MI455X (gfx1250, wave32).
// - All GEMMs use V_WMMA_F32_16X16X4_F32 (full fp32): workload is HBM-bound
//   (~0.5 GB streamed vs 36 GFLOP), so exact fp32 matrix ops cost nothing.
// - Gathers/scatters target xs/xd/agg (77 MB) -> resident in 192 MB L2.
// - Edge tile staged via GLOBAL_LOAD_ASYNC_TO_LDS_B128 (ASYNCcnt).
// - GELU uses hardware V_TANH_F32 when available.
// ---------------------------------------------------------------------------

#define N_NODES 50000
#define N_EDGES 800000
#define FEAT    128              // F_IN == H == F_OUT
#define E_DIM   64
#define MT      4                // row-tiles per wave in node GEMMs

typedef __attribute__((ext_vector_type(2))) float v2f;
typedef __attribute__((ext_vector_type(8))) float v8f;

__device__ __forceinline__ v8f wmma_f32_k4(v2f a, v2f b, v8f c) {
  // 8 args: (neg_a, A, neg_b, B, c_mod, C, reuse_a, reuse_b)
  return __builtin_amdgcn_wmma_f32_16x16x4_f32(
      false, a, false, b, (short)0, c, false, false);
}

// jax.nn.gelu default (approximate=True): tanh formulation.
// gfx1250 has a hardware tanh transcendental -> single V_TANH_F32.
__device__ __forceinline__ float gelu_tanh(float x) {
  const float c0 = 0.7978845608028654f;   // sqrt(2/pi)
  const float c1 = 0.044715f;
  float arg = c0 * (x + c1 * x * x * x);
#if __has_builtin(__builtin_amdgcn_tanhf)
  float t = __builtin_amdgcn_tanhf(arg);
#else
  float t = tanhf(arg);
#endif
  return 0.5f * x * (1.0f + t);
}

// ---------------------------------------------------------------------------
// C[M,128] = act(A[M,128] @ W[128,128] + bias)
// One wave owns MT row-tiles x one 16-col tile; 8 waves/block cover 128 cols.
// A-frag (f32 16x4): lane<16 -> K=k+0,k+1 ; lane>=16 -> K=k+2,k+3 ; M=lane&15.
// C-frag: VGPR r -> M = r + 8*(lane>=16), N = lane&15.
// Remainder tiles are clamped (redundant identical recompute, no divergence).
// ---------------------------------------------------------------------------
__global__ __launch_bounds__(256) void node_gemm_kernel(
    const float* __restrict__ A, const float* __restrict__ W,
    const float* __restrict__ bias, float* __restrict__ C,
    int nRowTiles, int applyGelu) {
  const int wave  = threadIdx.x >> 5;
  const int lane  = threadIdx.x & 31;
  const int khalf = lane >> 4;
  const int nlo   = lane & 15;
  const int col   = wave * 16 + nlo;

  int rt[MT];
#pragma unroll
  for (int i = 0; i < MT; ++i) {
    int t = blockIdx.x * MT + i;
    rt[i] = t < nRowTiles ? t : nRowTiles - 1;   // clamp remainder
  }

  v8f acc[MT];
#pragma unroll
  for (int i = 0; i < MT; ++i) acc[i] = (v8f){};

  for (int k = 0; k < FEAT; k += 4) {
    const int ka = k + 2 * khalf;
    v2f b;
    b.x = W[(size_t)ka * FEAT + col];
    b.y = W[(size_t)(ka + 1) * FEAT + col];
#pragma unroll
    for (int i = 0; i < MT; ++i) {
      v2f a = *(const v2f*)(A + (size_t)(rt[i] * 16 + nlo) * FEAT + ka);
      acc[i] = wmma_f32_k4(a, b, acc[i]);
    }
  }

  const float bv = bias[col];
#pragma unroll
  for (int i = 0; i < MT; ++i) {
#pragma unroll
    for (int r = 0; r < 8; ++r) {
      const int m = rt[i] * 16 + r + 8 * khalf;
      float v = acc[i][r] + bv;
      if (applyGelu) v = gelu_tanh(v);
      C[(size_t)m * FEAT + col] = v;
    }
  }
}

// ---------------------------------------------------------------------------
// xs = A @ Wsrc ; xd = A @ Wdst  (shared A-fragments, 2*MT accumulators)
// ---------------------------------------------------------------------------
__global__ __launch_bounds__(256) void node_dual_gemm_kernel(
    const float* __restrict__ A, const float* __restrict__ Wsrc,
    const float* __restrict__ Wdst, float* __restrict__ xs,
    float* __restrict__ xd, int nRowTiles) {
  const int wave  = threadIdx.x >> 5;
  const int lane  = threadIdx.x & 31;
  const int khalf = lane >> 4;
  const int nlo   = lane & 15;
  const int col   = wave * 16 + nlo;

  int rt[MT];
#pragma unroll
  for (int i = 0; i < MT; ++i) {
    int t = blockIdx.x * MT + i;
    rt[i] = t < nRowTiles ? t : nRowTiles - 1;
  }

  v8f accS[MT], accD[MT];
#pragma unroll
  for (int i = 0; i < MT; ++i) { accS[i] = (v8f){}; accD[i] = (v8f){}; }

  for (int k = 0; k < FEAT; k += 4) {
    const int ka = k + 2 * khalf;
    v2f bs, bd;
    bs.x = Wsrc[(size_t)ka * FEAT + col];
    bs.y = Wsrc[(size_t)(ka + 1) * FEAT + col];
    bd.x = Wdst[(size_t)ka * FEAT + col];
    bd.y = Wdst[(size_t)(ka + 1) * FEAT + col];
#pragma unroll
    for (int i = 0; i < MT; ++i) {
      v2f a = *(const v2f*)(A + (size_t)(rt[i] * 16 + nlo) * FEAT + ka);
      accS[i] = wmma_f32_k4(a, bs, accS[i]);
      accD[i] = wmma_f32_k4(a, bd, accD[i]);
    }
  }

#pragma unroll
  for (int i = 0; i < MT; ++i) {
#pragma unroll
    for (int r = 0; r < 8; ++r) {
      const int m = rt[i] * 16 + r + 8 * khalf;
      xs[(size_t)m * FEAT + col] = accS[i][r];
      xd[(size_t)m * FEAT + col] = accD[i][r];
    }
  }
}

// ---------------------------------------------------------------------------
// Per 16-edge tile: em = e[16,64] @ We[64,128] via WMMA (A staged in LDS via
// async-to-LDS loads), then fused in-fragment epilogue:
//   m = GELU(em + xs[src] + xd[dst] + b);  agg[dst] += m  (fp32 HW atomics)
// ---------------------------------------------------------------------------
__global__ __launch_bounds__(256) void edge_message_kernel(
    const float* __restrict__ efeat,   // [E, 64]
    const float* __restrict__ We,      // [64, 128]
    const float* __restrict__ bias,    // [128]
    const long long* __restrict__ srcIdx,
    const long long* __restrict__ dstIdx,
    const float* __restrict__ xs, const float* __restrict__ xd,
    float* __restrict__ agg) {
  __shared__ float etile[16 * E_DIM];  // 4 KB

  const long long ebase = (long long)blockIdx.x * 16;
  {
    // 256 lanes x 16B = 16x64 tile streamed straight into LDS (no VGPR
    // round-trip), tracked by ASYNCcnt.
    const float* g = efeat + (size_t)ebase * E_DIM + threadIdx.x * 4;
    unsigned ldsAddr =
        (unsigned)(uintptr_t)(etile + threadIdx.x * 4);
    unsigned long long gaddr = (unsigned long long)(uintptr_t)g;
    asm volatile("global_load_async_to_lds_b128 %0, %1, off"
                 :: "v"(ldsAddr), "v"(gaddr) : "memory");
    asm volatile("s_wait_asynccnt 0x0" ::: "memory");
  }
  __syncthreads();

  const int wave  = threadIdx.x >> 5;
  const int lane  = threadIdx.x & 31;
  const int khalf = lane >> 4;
  const int nlo   = lane & 15;
  const int col   = wave * 16 + nlo;

  const float* arow = etile + nlo * E_DIM;
  v8f acc = {};
#pragma unroll
  for (int k = 0; k < E_DIM; k += 4) {
    const int ka = k + 2 * khalf;
    v2f a = *(const v2f*)(arow + ka);   // ds_load_b64
    v2f b;
    b.x = We[(size_t)ka * FEAT + col];  // 32KB -> cache-resident
    b.y = We[(size_t)(ka + 1) * FEAT + col];
    acc = wmma_f32_k4(a, b, acc);
  }
  const float bv = bias[col];
#pragma unroll
  for (int r = 0; r < 8; ++r) {
    const long long e = ebase + r + 8 * khalf;
    const long long s = srcIdx[e];
    const long long d = dstIdx[e];
    float v = acc[r] + xs[(size_t)s * FEAT + col] +
              xd[(size_t)d * FEAT + col] + bv;
    v = gelu_tanh(v);
    __hip_atomic_fetch_add(&agg[(size_t)d * FEAT + col], v,
                           __ATOMIC_RELAXED, __HIP_MEMORY_SCOPE_AGENT);
  }
}

__global__ __launch_bounds__(256) void residual_add_kernel(
    float* __restrict__ h, const float* __restrict__ agg, int n) {
  int i = blockIdx.x * blockDim.x + threadIdx.x;
  if (i < n) h[i] += agg[i];
}

// ---------------------------------------------------------------------------
extern "C" void kernel_launch(void* const* d_in, const int* in_sizes, int n_in,
                              void* d_out, int out_size, void* d_ws,
                              size_t ws_size, hipStream_t stream) {
  (void)in_sizes; (void)n_in; (void)out_size; (void)ws_size;

  const float*     x        = (const float*)d_in[0];
  const long long* eidx     = (const long long*)d_in[1];   // int64 [2, E]
  const float*     efeat    = (const float*)d_in[2];
  const float*     ff1_W    = (const float*)d_in[3];
  const float*     ff1_b    = (const float*)d_in[4];
  const float*     mp1_Wsrc = (const float*)d_in[5];
  const float*     mp1_Wdst = (const float*)d_in[6];
  const float*     mp1_We   = (const float*)d_in[7];
  const float*     mp1_b    = (const float*)d_in[8];
  const float*     mp2_Wsrc = (const float*)d_in[9];
  const float*     mp2_Wdst = (const float*)d_in[10];
  const float*     mp2_We   = (const float*)d_in[11];
  const float*     mp2_b    = (const float*)d_in[12];
  const float*     ff2_W    = (const float*)d_in[13];
  const float*     ff2_b    = (const float*)d_in[14];
  float*           out      = (float*)d_out;

  const long long* srcIdx = eidx;              // edge_index[0]
  const long long* dstIdx = eidx + N_EDGES;    // edge_index[1]

  const size_t NF = (size_t)N_NODES * FEAT;    // 6.4M floats
  float* h   = (float*)d_ws;
  float* xs  = h + NF;
  float* xd  = xs + NF;
  float* agg = xd + NF;                        // total 102.4 MB

  const dim3 blk(256);
  const int nodeTiles = N_NODES / 16;                  // 3125 (exact)
  const int nodeBlks  = (nodeTiles + MT - 1) / MT;     // 782
  const int edgeTiles = N_EDGES / 16;                  // 50000 (exact)
  const int nElem     = (int)NF;

  // h = GELU(x @ ff1_W + b)
  node_gemm_kernel<<<nodeBlks, blk, 0, stream>>>(x, ff1_W, ff1_b, h,
                                                 nodeTiles, 1);

  // ---- message passing layer 1 ----
  node_dual_gemm_kernel<<<nodeBlks, blk, 0, stream>>>(h, mp1_Wsrc, mp1_Wdst,
                                                      xs, xd, nodeTiles);
  hipMemsetAsync(agg, 0, NF * sizeof(float), stream);
  edge_message_kernel<<<edgeTiles, blk, 0, stream>>>(efeat, mp1_We, mp1_b,
                                                     srcIdx, dstIdx, xs, xd,
                                                     agg);
  residual_add_kernel<<<(nElem + 255) / 256, blk, 0, stream>>>(h, agg, nElem);

  // ---- message passing layer 2 ----
  node_dual_gemm_kernel<<<nodeBlks, blk, 0, stream>>>(h, mp2_Wsrc, mp2_Wdst,
                                                      xs, xd, nodeTiles);
  hipMemsetAsync(agg, 0, NF * sizeof(float), stream);
  edge_message_kernel<<<edgeTiles, blk, 0, stream>>>(efeat, mp2_We, mp2_b,
                                                     srcIdx, dstIdx, xs, xd,
                                                     agg);
  residual_add_kernel<<<(nElem + 255) / 256, blk, 0, stream>>>(h, agg, nElem);

  // out = h @ ff2_W + b (no activation)
  node_gemm_kernel<<<nodeBlks, blk, 0, stream>>>(h, ff2_W, ff2_b, out,
                                                 nodeTiles, 0);
}